// Conv2D_67130338836622
// MI455X (gfx1250) — compile-verified
//
#include <hip/hip_runtime.h>
#include <stdint.h>

typedef float v2f __attribute__((ext_vector_type(2)));
typedef float v8f __attribute__((ext_vector_type(8)));

#if defined(__HIP_DEVICE_COMPILE__) && defined(__AMDGCN__)
#if !__has_builtin(__builtin_amdgcn_wmma_f32_16x16x4_f32)
#error "missing __builtin_amdgcn_wmma_f32_16x16x4_f32 on device toolchain"
#endif
#endif

#define IH 4096
#define IW 4096
#define OH 4090
#define OW 4090

// LDS halo tile: 70 input rows x 72 cols (71 needed, 72 for stride padding)
#define LROWS 70
#define LSTR  72

__global__ __launch_bounds__(256) void conv7x7_wmma_f32(
    const float* __restrict__ x, const float* __restrict__ wgt,
    float* __restrict__ out)
{
#if defined(__HIP_DEVICE_COMPILE__)
    __shared__ float Xs[LROWS * LSTR];

    const int tid  = threadIdx.x;
    const int lane = tid & 31;
    const int wave = tid >> 5;
    const int gx0  = blockIdx.x * 64;
    const int gy0  = blockIdx.y * 64;

    // ---------- stage 70x72 input halo tile into LDS via async copy ----------
    // Per-lane b32 async global->LDS (ASYNCcnt path). Clamped indexing keeps
    // addresses in-bounds; clamped lanes just rewrite the last element.
    for (int s = 0; s < LROWS * LSTR; s += 256) {
        int i = s + tid;
        if (i > LROWS * LSTR - 1) i = LROWS * LSTR - 1;
        int r = i / LSTR, c = i - r * LSTR;
        int gy = gy0 + r; if (gy > IH - 1) gy = IH - 1;
        int gx = gx0 + c; if (gx > IW - 1) gx = IW - 1;
        uint32_t lds_addr = (uint32_t)(uintptr_t)(&Xs[i]);
        uint64_t gaddr    = (uint64_t)(uintptr_t)(x + (size_t)gy * IW + gx);
        asm volatile("global_load_async_to_lds_b32 %0, %1, off"
                     :: "v"(lds_addr), "v"(gaddr) : "memory");
    }
    asm volatile("s_wait_asynccnt 0" ::: "memory");
    __syncthreads();

    // ---------- build A (weights) for the two K=4 WMMAs ----------
    // A is 16x8 (M=dy, K=dx). Rows 0..6 = W rows, row 7 = 0,
    // rows 8..14 duplicate rows 0..6 (half-replication trick), row 15 = 0.
    // A VGPR layout: lanes 0-15 hold K={0,1}, lanes 16-31 hold K={2,3}.
    const int  m      = lane & 15;
    const int  mm     = m & 7;            // duplicated dy index
    const bool hiHalf = (lane >= 16);
    const int  c0     = hiHalf ? 2 : 0;

    float w0 = (mm < 7 && (c0 + 0) < 7) ? wgt[mm * 7 + c0 + 0] : 0.0f;
    float w1 = (mm < 7 && (c0 + 1) < 7) ? wgt[mm * 7 + c0 + 1] : 0.0f;
    float w4 = (mm < 7 && (c0 + 4) < 7) ? wgt[mm * 7 + c0 + 4] : 0.0f;
    float w5 = (mm < 7 && (c0 + 5) < 7) ? wgt[mm * 7 + c0 + 5] : 0.0f;
    v2f A0 = { w0, w1 };   // dx block 0..3
    v2f A1 = { w4, w5 };   // dx block 4..7 (col 7 zero-padded)

    const v8f zero8 = {0.f, 0.f, 0.f, 0.f, 0.f, 0.f, 0.f, 0.f};
    const int n = lane & 15;

    // ---------- each wave computes two 16x16 output tiles ----------
    for (int t = wave; t < 16; t += 8) {
        const int ty = t >> 2, tx = t & 3;
        const int xb = tx * 16;     // tile col base in LDS
        const int rb = ty * 16;     // first input row (LDS-local)

        v8f acc = zero8;

        #pragma unroll
        for (int rr = 0; rr < 22; ++rr) {
            // B is 4x16 (K=dx, N=x): per-lane band from one LDS input row.
            // B VGPR layout mirrors A: lanes 0-15 K={0,1}, lanes 16-31 K={2,3}.
            const float* p = &Xs[(rb + rr) * LSTR + xb + n + c0];
            v2f B0 = { p[0], p[1] };
            v2f B1 = { p[4], p[5] };

            v8f pj = __builtin_amdgcn_wmma_f32_16x16x4_f32(
                false, A0, false, B0, (short)0, zero8, false, false);
            pj = __builtin_amdgcn_wmma_f32_16x16x4_f32(
                false, A1, false, B1, (short)0, pj, false, false);

            // Diagonal reduction: O[y=rr-j, x] += P[rr, x, dy=j].
            // pj[j] holds P[dy=j] identically in both lane halves, so a
            // half-select (cndmask) routes it into the right accumulator half.
            #pragma unroll
            for (int j = 0; j < 7; ++j) {
                const int y = rr - j;
                if (y >= 0 && y < 16) {
                    float v = (((y >> 3) == (hiHalf ? 1 : 0))) ? pj[j] : 0.0f;
                    acc[y & 7] += v;
                }
            }
        }

        // ---------- store 16x16 tile (C layout: vgpr i = rows i, i+8) ----------
        const int ox = gx0 + xb + n;
        const int oyb = gy0 + rb + (hiHalf ? 8 : 0);
        #pragma unroll
        for (int i2 = 0; i2 < 8; ++i2) {
            const int oy = oyb + i2;
            if (oy < OH && ox < OW)
                out[(size_t)oy * OW + ox] = acc[i2];
        }
    }
#endif // __HIP_DEVICE_COMPILE__
}

extern "C" void kernel_launch(void* const* d_in, const int* in_sizes, int n_in,
                              void* d_out, int out_size, void* d_ws, size_t ws_size,
                              hipStream_t stream) {
    const float* x = (const float*)d_in[0];   // 4096*4096 f32
    const float* w = (const float*)d_in[1];   // 7*7 f32
    float* out = (float*)d_out;               // 4090*4090 f32
    dim3 grid((OW + 63) / 64, (OH + 63) / 64);
    conv7x7_wmma_f32<<<grid, dim3(256), 0, stream>>>(x, w, out);
}